// RegionFeatureExtractorBatch_10617159155853
// MI455X (gfx1250) — compile-verified
//
#include <hip/hip_runtime.h>
#include <math.h>

typedef float v2f __attribute__((ext_vector_type(2)));
typedef float v8f __attribute__((ext_vector_type(8)));

#define Bc 8
#define Cc 256
#define Hf 160
#define Wf 160
#define Nb 128
#define Sg 7
#define HP 161
#define WP 161
#define PLANE (HP * WP)   /* 25921 */
#define FPLANE (Hf * Wf)  /* 25600 */

// ---------------------------------------------------------------------------
// Kernel 1: projection GEMM via V_WMMA_F32_16X16X4_F32.
// Whole proj_w (256 KB) staged in LDS (CDNA5 WGP has 320 KB LDS).
// Block = 256 threads = 8 waves. Wave handles 16 pixels x 128 out-channels.
// Writes results into the interior of the padded integral buffer I.
// ---------------------------------------------------------------------------
__global__ __launch_bounds__(256) void proj_gemm_wmma(
    const float* __restrict__ fm, const float* __restrict__ Wm,
    const float* __restrict__ bias, float* __restrict__ I)
{
    __shared__ __align__(16) float smW[Cc * Cc];  // 256 KB
    __shared__ float smB[Cc];

    const int tid = threadIdx.x;

    // Cooperative load of proj_w into LDS (float4, coalesced)
    {
        const float4* src = (const float4*)Wm;
        float4* dst = (float4*)smW;
        #pragma unroll 4
        for (int i = tid; i < (Cc * Cc) / 4; i += 256) dst[i] = src[i];
        smB[tid] = bias[tid];
    }
    __syncthreads();

    const int b    = blockIdx.x / 400;   // 400 blocks per batch image
    const int blk  = blockIdx.x % 400;
    const int wave = tid >> 5;
    const int lane = tid & 31;
    const int lhalf = lane >> 4;         // 0: K0/K1 ; 1: K2/K3 (A & B frag halves)
    const int lmod  = lane & 15;         // M row (A) / N col (B)

    const int mbase   = (wave & 1) * 128;          // 8 M-tiles of 16 -> 128 channels
    const int tileIdx = blk * 4 + (wave >> 1);     // 0..1599, 16-pixel tiles
    const int h  = tileIdx / 10;                   // 10 tiles per 160-wide row
    const int w0 = (tileIdx % 10) * 16;

    const float* fmb = fm + (size_t)b * Cc * FPLANE + (size_t)h * Wf + w0 + lmod;

    v8f acc[8];
    #pragma unroll
    for (int t = 0; t < 8; ++t) acc[t] = (v8f){0.f, 0.f, 0.f, 0.f, 0.f, 0.f, 0.f, 0.f};

    for (int k0 = 0; k0 < Cc; k0 += 4) {
        const int kk = k0 + 2 * lhalf;
        // B fragment: B[K,N] = fm[b, kk + {0,1}, pixel]; lanes 0-15 K0/K1, 16-31 K2/K3
        v2f bfrag;
        bfrag.x = fmb[(size_t)kk * FPLANE];
        bfrag.y = fmb[(size_t)(kk + 1) * FPLANE];
        #pragma unroll
        for (int t = 0; t < 8; ++t) {
            const int row = mbase + t * 16 + lmod;
            // A fragment from LDS: proj_w[row][kk], proj_w[row][kk+1] (ds_load_b64)
            v2f afrag = *(const v2f*)&smW[row * Cc + kk];
            acc[t] = __builtin_amdgcn_wmma_f32_16x16x4_f32(
                false, afrag, false, bfrag, (short)0, acc[t], false, false);
        }
    }

    // Epilogue: add bias, store into padded I interior at (h+1, w+1)
    const size_t outcol = (size_t)(h + 1) * WP + (size_t)(w0 + lmod + 1);
    #pragma unroll
    for (int t = 0; t < 8; ++t) {
        #pragma unroll
        for (int r = 0; r < 8; ++r) {
            const int row = mbase + t * 16 + r + 8 * lhalf;   // C/D layout: lanes>=16 hold M+8
            const float v = acc[t][r] + smB[row];
            I[((size_t)b * Cc + row) * PLANE + outcol] = v;
        }
    }
}

// ---------------------------------------------------------------------------
// Kernel 2: in-place row prefix sum (along w). One wave32 per row.
// Row h==0 is the zero pad row; column 0 of every row is zeroed.
// ---------------------------------------------------------------------------
__global__ __launch_bounds__(256) void row_scan(float* __restrict__ I)
{
    const int wave = threadIdx.x >> 5;
    const int lane = threadIdx.x & 31;
    const int gw = blockIdx.x * 8 + wave;        // grid sized exactly: Bc*Cc*HP waves
    const int plane = gw / HP;
    const int h = gw % HP;
    float* row = I + (size_t)plane * PLANE + (size_t)h * WP;

    if (h == 0) {
        for (int c = lane; c < WP; c += 32) row[c] = 0.f;
        return;
    }
    if (lane == 0) row[0] = 0.f;

    float carry = 0.f;
    #pragma unroll
    for (int ch = 0; ch < 5; ++ch) {             // 160 = 5 chunks of 32
        const int c = 1 + ch * 32 + lane;
        float v = row[c];
        #pragma unroll
        for (int d = 1; d < 32; d <<= 1) {
            float n = __shfl_up(v, d, 32);
            if (lane >= d) v += n;
        }
        v += carry;
        row[c] = v;
        carry = __shfl(v, 31, 32);
    }
}

// ---------------------------------------------------------------------------
// Kernel 3: in-place column prefix sum (along h). One thread per column;
// consecutive threads hit consecutive addresses each step (coalesced).
// ---------------------------------------------------------------------------
__global__ __launch_bounds__(256) void col_scan(float* __restrict__ I)
{
    const int id = blockIdx.x * 256 + threadIdx.x;
    if (id >= Bc * Cc * WP) return;
    const int plane = id / WP;
    const int w = id % WP;
    float* base = I + (size_t)plane * PLANE + w;
    float acc = 0.f;
    for (int h = 0; h < HP; ++h) {
        acc += base[(size_t)h * WP];
        base[(size_t)h * WP] = acc;
    }
}

// ---------------------------------------------------------------------------
// Kernel 4: region pooling. Block per (b, box); thread = channel.
// Corner gathers from I; result staged in LDS, then written as one fully
// coalesced contiguous 12544-float block. Thread 0 writes the area ratio.
// ---------------------------------------------------------------------------
__global__ __launch_bounds__(256) void region_pool(
    const float* __restrict__ I, const float* __restrict__ bboxes,
    float* __restrict__ outRF, float* __restrict__ outAR)
{
    __shared__ float smOut[Cc * Sg * Sg];   // 12544 floats = ~50 KB
    __shared__ int smRS[Sg], smRE[Sg], smCS[Sg], smCE[Sg];

    const int bn = blockIdx.x;              // 0..1023
    const int b = bn / Nb;
    const int tid = threadIdx.x;

    if (tid == 0) {
        const float* bb = bboxes + (size_t)bn * 4;
        const float x1f = bb[0], y1f = bb[1], x2f = bb[2], y2f = bb[3];
        int x1 = (int)floorf(x1f * 0.25f); x1 = min(max(x1, 0), Wf - 1);
        int y1 = (int)floorf(y1f * 0.25f); y1 = min(max(y1, 0), Hf - 1);
        int x2 = (int)floorf(x2f * 0.25f); x2 = min(max(x2, x1 + 1), Wf);
        int y2 = (int)floorf(y2f * 0.25f); y2 = min(max(y2, y1 + 1), Hf);
        const int Lx = x2 - x1, Ly = y2 - y1;
        #pragma unroll
        for (int i = 0; i < Sg; ++i) {
            smRS[i] = y1 + (i * Ly) / Sg;
            smRE[i] = y1 + ((i + 1) * Ly + Sg - 1) / Sg;
            smCS[i] = x1 + (i * Lx) / Sg;
            smCE[i] = x1 + ((i + 1) * Lx + Sg - 1) / Sg;
        }
        outAR[bn] = (x2f - x1f) * (y2f - y1f);
    }
    __syncthreads();

    const int c = tid;
    const float* P = I + ((size_t)b * Cc + c) * PLANE;
    for (int i = 0; i < Sg; ++i) {
        const int r0 = smRS[i] * WP;
        const int r1 = smRE[i] * WP;
        const float dh = (float)(smRE[i] - smRS[i]);
        #pragma unroll
        for (int j = 0; j < Sg; ++j) {
            const int c0 = smCS[j], c1 = smCE[j];
            const float tl = P[r0 + c0];
            const float tr = P[r0 + c1];
            const float bl = P[r1 + c0];
            const float br = P[r1 + c1];
            const float cnt = dh * (float)(c1 - c0);
            smOut[c * (Sg * Sg) + i * Sg + j] = (br - tr - bl + tl) / cnt;
        }
    }
    __syncthreads();

    float* dst = outRF + (size_t)bn * Cc * (Sg * Sg);
    for (int k = tid; k < Cc * Sg * Sg; k += 256) dst[k] = smOut[k];
}

// ---------------------------------------------------------------------------
extern "C" void kernel_launch(void* const* d_in, const int* in_sizes, int n_in,
                              void* d_out, int out_size, void* d_ws, size_t ws_size,
                              hipStream_t stream)
{
    const float* fm     = (const float*)d_in[0];  // [8,256,160,160]
    const float* bboxes = (const float*)d_in[1];  // [8,128,4]
    const float* Wm     = (const float*)d_in[2];  // [256,256]
    const float* bias   = (const float*)d_in[3];  // [256]
    (void)in_sizes; (void)n_in; (void)out_size; (void)ws_size;

    float* I = (float*)d_ws;                      // [8,256,161,161] = ~203 MB
    float* outRF = (float*)d_out;                 // [8,128,256,7,7]
    float* outAR = outRF + (size_t)Bc * Nb * Cc * Sg * Sg;  // [8,128]

    // 1) projection GEMM (WMMA) -> padded I interior
    proj_gemm_wmma<<<dim3(Bc * 400), dim3(256), 0, stream>>>(fm, Wm, bias, I);
    // 2) row prefix sums (+ pad zeroing): Bc*Cc*HP waves, 8 waves/block
    row_scan<<<dim3((Bc * Cc * HP) / 8), dim3(256), 0, stream>>>(I);
    // 3) column prefix sums
    col_scan<<<dim3((Bc * Cc * WP + 255) / 256), dim3(256), 0, stream>>>(I);
    // 4) box pooling + area ratios
    region_pool<<<dim3(Bc * Nb), dim3(256), 0, stream>>>(I, bboxes, outRF, outAR);
}